// WindowAttention_82824149336760
// MI455X (gfx1250) — compile-verified
//
#include <hip/hip_runtime.h>

typedef __attribute__((ext_vector_type(2))) float v2f;
typedef __attribute__((ext_vector_type(8))) float v8f;

#define SEQ   144
#define CH    512
#define CH3   1536
#define NH    16
#define DH    32
#define NWIN  64
#define BNTOT 256

// LDS strides padded for conflict-free WMMA fragment access (64 banks, wave32):
// A-fragment reads addr = m*S + k0 + {0..3}; need {m*S mod 64} to be 16 distinct
// multiples of 4 -> S mod 64 in {4,20,36,...} works (gcd(S mod 64, 64) == 4).
#define QKV_STRIDE 100   // 100 mod 64 = 36 : ok
#define S_STRIDE   148   // 148 mod 64 = 20 : ok
#define XS_STRIDE  68    //  68 mod 64 =  4 : ok
#define WS_STRIDE  104   // B reads: 2*104 mod 64 = 16 -> lane halves disjoint

__device__ __forceinline__ v8f wmma4(v2f a, v2f b, v8f c) {
  // D = A(16x4 f32) x B(4x16 f32) + C(16x16 f32)
  return __builtin_amdgcn_wmma_f32_16x16x4_f32(
      /*neg_a=*/false, a, /*neg_b=*/false, b,
      /*c_mod=*/(short)0, c, /*reuse_a=*/false, /*reuse_b=*/false);
}

// ---------------------------------------------------------------------------
// Kernel 1: fused QKV projection + masked softmax attention for one
// (batch-window bn, head h). Writes attention output in (Bn, N, C) layout
// directly into `out` (later overwritten in-place by the projection kernel).
// ---------------------------------------------------------------------------
__global__ __launch_bounds__(256) void fused_window_attention(
    const float* __restrict__ x, const int* __restrict__ mask,
    const float* __restrict__ Wqkv, const float* __restrict__ bqkv,
    float* __restrict__ out) {
  __shared__ float ldsQKV[SEQ * QKV_STRIDE];  // Q | K | V in cols 0..31|32..63|64..95
  __shared__ float ldsS[SEQ * S_STRIDE];      // scores / softmax probs (+ staging overlay)

  const int h    = blockIdx.x;
  const int bn   = blockIdx.y;
  const int wdx  = bn & (NWIN - 1);     // window index = bn % nW
  const int tid  = threadIdx.x;
  const int lane = tid & 31;
  const int wv   = tid >> 5;            // 8 waves
  const int lm   = lane & 15;           // M (A) / N (B,C,D) position within tile
  const int kh   = (lane >> 4) << 1;    // K sub-offset: lanes 16..31 hold K+2,K+3
  const int rhi  = (lane >> 4) << 3;    // C/D rows: lanes 16..31 hold M+8

  // Staging buffers for step 1 overlay the (not-yet-used) score buffer.
  float* xs = ldsS;                     // x chunk:    [144][XS_STRIDE]
  float* ws = ldsS + SEQ * XS_STRIDE;   // Wqkv chunk: [64][WS_STRIDE] (96 cols used)

  const v8f vzero = {0.f, 0.f, 0.f, 0.f, 0.f, 0.f, 0.f, 0.f};

  // ---- Step 1: [Q|K|V](144x96) = x[bn](144x512) @ Wqkv[:, head cols] + bias ----
  // 54 output tiles (9 M-tiles x 6 N-tiles) round-robined over 8 waves (<=7 each).
  v8f acc1[7];
  #pragma unroll
  for (int i = 0; i < 7; ++i) acc1[i] = vzero;

  for (int kc = 0; kc < CH; kc += 64) {
    for (int i = tid; i < SEQ * 16; i += 256) {           // 144 rows x 16 float4
      const int m = i >> 4, j = (i & 15) << 2;
      const float4 v = *(const float4*)(x + ((size_t)bn * SEQ + m) * CH + kc + j);
      *(float4*)(xs + m * XS_STRIDE + j) = v;
    }
    for (int i = tid; i < 64 * 24; i += 256) {            // 64 rows x 24 float4 (96 cols)
      const int kr = i / 24, cc = (i % 24) << 2;
      const int qsel = cc >> 5, c = cc & 31;
      const float4 v = *(const float4*)(Wqkv + (size_t)(kc + kr) * CH3 + qsel * CH + h * DH + c);
      *(float4*)(ws + kr * WS_STRIDE + cc) = v;
    }
    __syncthreads();
    #pragma unroll
    for (int i = 0; i < 7; ++i) {
      const int t = i * 8 + wv;                           // wave-uniform guard
      if (t < 54) {
        const int tm = t / 6, tn = t % 6;
        const float* arow = xs + (tm * 16 + lm) * XS_STRIDE;
        const float* bcol = ws + tn * 16 + lm;
        #pragma unroll
        for (int k0 = 0; k0 < 64; k0 += 4) {
          const int ko = k0 + kh;
          const v2f a = *(const v2f*)(arow + ko);
          v2f b; b.x = bcol[ko * WS_STRIDE]; b.y = bcol[(ko + 1) * WS_STRIDE];
          acc1[i] = wmma4(a, b, acc1[i]);
        }
      }
    }
    __syncthreads();
  }

  #pragma unroll
  for (int i = 0; i < 7; ++i) {
    const int t = i * 8 + wv;
    if (t < 54) {
      const int tm = t / 6, tn = t % 6;
      const int cc = tn * 16 + lm;
      const int qsel = cc >> 5, c = cc & 31;
      const float bias = bqkv[qsel * CH + h * DH + c];
      #pragma unroll
      for (int r = 0; r < 8; ++r)
        ldsQKV[(tm * 16 + r + rhi) * QKV_STRIDE + cc] = acc1[i][r] + bias;
    }
  }
  __syncthreads();

  // ---- Step 2: S = (Q @ K^T) * scale + mask_bias ; 81 tiles over 8 waves ----
  const float scale = 0.17677669529663687f;  // 1/sqrt(32)
  #pragma unroll
  for (int i = 0; i < 11; ++i) {
    const int t = i * 8 + wv;
    if (t < 81) {
      const int tm = t / 9, tn = t % 9;
      const float* qrow = ldsQKV + (tm * 16 + lm) * QKV_STRIDE;       // A: Q row-major
      const float* krow = ldsQKV + (tn * 16 + lm) * QKV_STRIDE + DH;  // B[k][n]=K[n][k]: row read
      v8f acc = vzero;
      #pragma unroll
      for (int k0 = 0; k0 < DH; k0 += 4) {
        const int ko = k0 + kh;
        acc = wmma4(*(const v2f*)(qrow + ko), *(const v2f*)(krow + ko), acc);
      }
      const int gn = tn * 16 + lm;
      #pragma unroll
      for (int r = 0; r < 8; ++r) {
        const int gm = tm * 16 + r + rhi;
        float sc = acc[r] * scale;
        if (mask[((size_t)wdx * SEQ + gm) * SEQ + gn] == 0) sc = -1e30f;
        ldsS[gm * S_STRIDE + gn] = sc;
      }
    }
  }
  __syncthreads();

  // ---- Step 3: row softmax (144 rows; one thread per row) ----
  if (tid < SEQ) {
    float* row = ldsS + tid * S_STRIDE;
    float mx = -3.4e38f;
    for (int j = 0; j < SEQ; ++j) mx = fmaxf(mx, row[j]);
    float sum = 0.f;
    for (int j = 0; j < SEQ; ++j) { const float e = __expf(row[j] - mx); row[j] = e; sum += e; }
    const float inv = 1.0f / sum;
    for (int j = 0; j < SEQ; ++j) row[j] *= inv;
  }
  __syncthreads();

  // ---- Step 4: O = P(144x144) @ V(144x32) ; 18 tiles over 8 waves ----
  #pragma unroll
  for (int i = 0; i < 3; ++i) {
    const int t = i * 8 + wv;
    if (t < 18) {
      const int tm = t >> 1, tn = t & 1;
      const float* prow = ldsS + (tm * 16 + lm) * S_STRIDE;
      const float* vcol = ldsQKV + 2 * DH + tn * 16 + lm;
      v8f acc = vzero;
      #pragma unroll
      for (int k0 = 0; k0 < SEQ; k0 += 4) {
        const int ko = k0 + kh;
        const v2f a = *(const v2f*)(prow + ko);
        v2f b; b.x = vcol[ko * QKV_STRIDE]; b.y = vcol[(ko + 1) * QKV_STRIDE];
        acc = wmma4(a, b, acc);
      }
      const int gcol = h * DH + tn * 16 + lm;
      #pragma unroll
      for (int r = 0; r < 8; ++r)
        out[((size_t)bn * SEQ + tm * 16 + r + rhi) * CH + gcol] = acc[r];
    }
  }
}

// ---------------------------------------------------------------------------
// Kernel 2: in-place output projection. Each block owns 32 rows of `io`:
// stages them fully (all K=512) into LDS first, then overwrites with
// out = A @ Wproj + bproj. No workspace needed.
// ---------------------------------------------------------------------------
#define PA_STRIDE 516   // 516 mod 64 = 4 : conflict-free A-fragment reads
#define PB_STRIDE 520   // 2*520 mod 64 = 16 : lane halves disjoint for B reads

__global__ __launch_bounds__(256) void proj_gemm(
    const float* __restrict__ Wproj, const float* __restrict__ bproj,
    float* __restrict__ io) {
  __shared__ float as_[32 * PA_STRIDE];   // 32 rows x 512 K
  __shared__ float bs_[32 * PB_STRIDE];   // 32 K x 512 cols chunk
  const int tid  = threadIdx.x;
  const int lane = tid & 31;
  const int wv   = tid >> 5;
  const int lm   = lane & 15;
  const int kh   = (lane >> 4) << 1;
  const int rhi  = (lane >> 4) << 3;
  const size_t rowBase = (size_t)blockIdx.x * 32;
  const int colBase = wv * 64;            // wave owns a 64-column strip

  for (int i = tid; i < 32 * 128; i += 256) {            // 32 rows x 128 float4
    const int m = i >> 7, j = (i & 127) << 2;
    const float4 v = *(const float4*)(io + (rowBase + m) * CH + j);
    *(float4*)(as_ + m * PA_STRIDE + j) = v;
  }

  const v8f vzero = {0.f, 0.f, 0.f, 0.f, 0.f, 0.f, 0.f, 0.f};
  v8f acc[2][4];
  #pragma unroll
  for (int im = 0; im < 2; ++im)
    #pragma unroll
    for (int in = 0; in < 4; ++in) acc[im][in] = vzero;

  __syncthreads();

  for (int kc = 0; kc < CH; kc += 32) {
    for (int i = tid; i < 32 * 128; i += 256) {          // Wproj chunk: 32 K x 512 cols
      const int kr = i >> 7, j = (i & 127) << 2;
      const float4 v = *(const float4*)(Wproj + (size_t)(kc + kr) * CH + j);
      *(float4*)(bs_ + kr * PB_STRIDE + j) = v;
    }
    __syncthreads();
    #pragma unroll
    for (int k0 = 0; k0 < 32; k0 += 4) {
      const int ko = k0 + kh;
      v2f a[2], b[4];
      #pragma unroll
      for (int im = 0; im < 2; ++im)
        a[im] = *(const v2f*)(as_ + (im * 16 + lm) * PA_STRIDE + kc + ko);
      #pragma unroll
      for (int in = 0; in < 4; ++in) {
        const float* bp = bs_ + colBase + in * 16 + lm;
        b[in].x = bp[ko * PB_STRIDE];
        b[in].y = bp[(ko + 1) * PB_STRIDE];
      }
      #pragma unroll
      for (int im = 0; im < 2; ++im)
        #pragma unroll
        for (int in = 0; in < 4; ++in)
          acc[im][in] = wmma4(a[im], b[in], acc[im][in]);
    }
    __syncthreads();
  }

  #pragma unroll
  for (int im = 0; im < 2; ++im) {
    #pragma unroll
    for (int in = 0; in < 4; ++in) {
      const int gcol = colBase + in * 16 + lm;
      const float bias = bproj[gcol];
      #pragma unroll
      for (int r = 0; r < 8; ++r)
        io[(rowBase + im * 16 + r + rhi) * CH + gcol] = acc[im][in][r] + bias;
    }
  }
}

extern "C" void kernel_launch(void* const* d_in, const int* in_sizes, int n_in,
                              void* d_out, int out_size, void* d_ws, size_t ws_size,
                              hipStream_t stream) {
  (void)in_sizes; (void)n_in; (void)out_size; (void)d_ws; (void)ws_size;
  const float* x     = (const float*)d_in[0];
  const int*   mask  = (const int*)d_in[1];
  const float* Wqkv  = (const float*)d_in[2];
  const float* bqkv  = (const float*)d_in[3];
  const float* Wproj = (const float*)d_in[4];
  const float* bproj = (const float*)d_in[5];
  float* out = (float*)d_out;

  // 4096 blocks: one per (head, batch-window)
  fused_window_attention<<<dim3(NH, BNTOT), 256, 0, stream>>>(x, mask, Wqkv, bqkv, out);
  // 1152 blocks: 32 output rows each, in-place over `out`
  proj_gemm<<<dim3((BNTOT * SEQ) / 32), 256, 0, stream>>>(Wproj, bproj, out);
}